// GNN2_test_54949811585066
// MI455X (gfx1250) — compile-verified
//
#include <hip/hip_runtime.h>

#define N_USER 6144
#define N_ITEM 6144
#define NTOT   (N_USER + N_ITEM)   // 12288
#define D      64
#define BATCH  2048
#define NNEG   4

#define KT   64                    // K-tile staged through LDS (double buffered)
#define APAD 4                     // As row stride 68 floats (float4-aligned, conflict-free)
#define BPAD 4                     // Bs row stride 132 floats

typedef float v2f __attribute__((ext_vector_type(2)));
typedef float v8f __attribute__((ext_vector_type(8)));
typedef int   v4i __attribute__((ext_vector_type(4)));

// ---- CDNA5 async global->LDS path (ASYNCcnt), with sync fallback ----
#if defined(__has_builtin)
#  if __has_builtin(__builtin_amdgcn_global_load_async_to_lds_b128) && \
      __has_builtin(__builtin_amdgcn_s_wait_asynccnt)
#    define USE_ASYNC 1
#  endif
#endif
#ifndef USE_ASYNC
#  define USE_ASYNC 0
#endif

#if USE_ASYNC
// builtin signature (from clang): (v4i addrspace(1)*, v4i addrspace(3)*, imm offset, imm cpol)
typedef __attribute__((address_space(1))) v4i* gptr_b128;
typedef __attribute__((address_space(3))) v4i* lptr_b128;

__device__ __forceinline__ void cp_async_b128(const float* g, float* l) {
    __builtin_amdgcn_global_load_async_to_lds_b128(
        (gptr_b128)(const_cast<float*>(g)),
        (lptr_b128)l,
        0, 0);
}
#endif

// ---------------- phase 1: flag + concatenated B panel ----------------

__global__ void zero_flag_kernel(float* __restrict__ flagf) {
    int i = blockIdx.x * blockDim.x + threadIdx.x;
    if (i < NTOT) flagf[i] = 0.0f;
}

__global__ void scatter_flag_kernel(const int* __restrict__ users,
                                    const int* __restrict__ pos_items,
                                    float* __restrict__ flagf) {
    int t = blockIdx.x * blockDim.x + threadIdx.x;
    if (t == 0) {
        flagf[users[0]] = 1.0f;
    } else if (t <= BATCH) {
        flagf[N_USER + pos_items[t - 1]] = 1.0f;
    }
}

// Bcat[r][0:64] = flag[r]*ego[r] ; Bcat[r][64:128] = (1-flag[r])*ego[r]
__global__ void build_bcat_kernel(const float* __restrict__ user_emb,
                                  const float* __restrict__ item_emb,
                                  const float* __restrict__ flagf,
                                  float* __restrict__ bcat) {
    int idx = blockIdx.x * blockDim.x + threadIdx.x;   // over NTOT*D
    if (idx >= NTOT * D) return;
    int r = idx / D;
    int d = idx - r * D;
    float e = (r < N_USER) ? user_emb[(size_t)r * D + d]
                           : item_emb[(size_t)(r - N_USER) * D + d];
    float f = flagf[r];
    bcat[(size_t)r * (2 * D) + d]     = f * e;
    bcat[(size_t)r * (2 * D) + D + d] = (1.0f - f) * e;
}

// ---------------- phase 2: WMMA GEMM + fused mask-combine + light_out ----------------
// grid = NTOT/16 blocks of 256 threads (8 waves). Each block: 16 rows x 128 cols.
// wave w owns output columns [w*16, w*16+16). Double-buffered async LDS staging.

__global__ __launch_bounds__(256) void gemm_light_kernel(
    const float* __restrict__ A,        // norm_adj [NTOT][NTOT]
    const float* __restrict__ bcat,     // [NTOT][128]
    const float* __restrict__ flagf,    // [NTOT]
    const float* __restrict__ user_emb, // [N_USER][64]
    const float* __restrict__ item_emb, // [N_ITEM][64]
    float* __restrict__ light)          // [NTOT][64]
{
    __shared__ float As[2][16][KT + APAD];    // 2 x 16 x 64 A slabs
    __shared__ float Bs[2][KT][128 + BPAD];   // 2 x 64 x 128 B slabs
    __shared__ float Cs[16][2 * D];           // 16 x 128 result tile

    const int tid  = threadIdx.x;
    const int wave = tid >> 5;      // 0..7 -> column tile
    const int lane = tid & 31;
    const int half = lane >> 4;     // lane half selects K pair (ISA 16x4 layout)
    const int l16  = lane & 15;     // M (for A) / N (for B)
    const int row0 = blockIdx.x * 16;

    v8f acc = {};

    auto stage = [&](int k0, int buf) {
        // A slab: 16 rows x KT cols -> 16x(KT/4)=256 float4 chunks, 1 per thread
        for (int i = tid; i < 16 * (KT / 4); i += 256) {
            int r  = i >> 4;            // KT/4 = 16 float4 per row
            int c4 = i & 15;
            const float* g = A + (size_t)(row0 + r) * NTOT + k0 + c4 * 4;
#if USE_ASYNC
            cp_async_b128(g, &As[buf][r][c4 * 4]);
#else
            *(float4*)&As[buf][r][c4 * 4] = *(const float4*)g;
#endif
        }
        // B slab: KT rows x 128 cols -> KT*32 float4 chunks, 8 per thread (L2-resident)
        for (int i = tid; i < KT * (128 / 4); i += 256) {
            int r  = i >> 5;
            int c4 = i & 31;
            const float* g = bcat + (size_t)(k0 + r) * 128 + c4 * 4;
#if USE_ASYNC
            cp_async_b128(g, &Bs[buf][r][c4 * 4]);
#else
            *(float4*)&Bs[buf][r][c4 * 4] = *(const float4*)g;
#endif
        }
    };

    int cur = 0;
    stage(0, 0);
#if USE_ASYNC
    __builtin_amdgcn_s_wait_asynccnt(0);
#endif
    __syncthreads();

    for (int k0 = 0; k0 < NTOT; k0 += KT) {
        int nk = k0 + KT;
        if (nk < NTOT) stage(nk, cur ^ 1);   // overlap next-tile DMA with compute

        // 16 WMMA steps of K=4 each on current buffer
        #pragma unroll
        for (int kk = 0; kk < KT; kk += 4) {
            v2f a, b;
            // A 16x4 layout: lanes 0-15 -> K=0,1 ; lanes 16-31 -> K=2,3
            a.x = As[cur][l16][kk + half * 2];
            a.y = As[cur][l16][kk + half * 2 + 1];
            // B 4x16 layout (transpose-symmetric): N = l16, K = half*2 + vgpr
            b.x = Bs[cur][kk + half * 2][wave * 16 + l16];
            b.y = Bs[cur][kk + half * 2 + 1][wave * 16 + l16];
            acc = __builtin_amdgcn_wmma_f32_16x16x4_f32(
                /*neg_a=*/false, a, /*neg_b=*/false, b,
                /*c_mod=*/(short)0, acc, /*reuse_a=*/false, /*reuse_b=*/false);
        }

#if USE_ASYNC
        __builtin_amdgcn_s_wait_asynccnt(0);  // next tile landed in LDS
#endif
        __syncthreads();                       // everyone done reading cur + loads visible
        cur ^= 1;
    }

    // spill C tile to LDS: C/D layout -> VGPR j holds row (j + half*8), col l16
    #pragma unroll
    for (int j = 0; j < 8; ++j)
        Cs[j + half * 8][wave * 16 + l16] = acc[j];
    __syncthreads();

    // combine halves with flag[r], fuse (ego + 3*prop)/4, write light rows
    for (int i = tid; i < 16 * D; i += 256) {
        int r  = i >> 6;
        int d  = i & 63;
        int gr = row0 + r;
        float p = Cs[r][d] + flagf[gr] * Cs[r][D + d];
        float e = (gr < N_USER) ? user_emb[(size_t)gr * D + d]
                                : item_emb[(size_t)(gr - N_USER) * D + d];
        light[(size_t)gr * D + d] = (e + 3.0f * p) * 0.25f;
    }
}

// ---------------- phase 3: gathers into d_out ----------------
// d_out = [u_g (B x 64) | pos_i (B x 64) | neg_i (4B x 64)] flat

__global__ void gather_out_kernel(const float* __restrict__ light,
                                  const int* __restrict__ users,
                                  const int* __restrict__ pos_items,
                                  const int* __restrict__ neg_items,
                                  float* __restrict__ out) {
    int idx = blockIdx.x * blockDim.x + threadIdx.x;   // over 6*BATCH*D
    if (idx >= (2 + NNEG) * BATCH * D) return;
    int t = idx / D;
    int d = idx - t * D;
    int src;
    if (t < BATCH)            src = users[t];
    else if (t < 2 * BATCH)   src = N_USER + pos_items[t - BATCH];
    else                      src = N_USER + neg_items[t - 2 * BATCH];
    out[idx] = light[(size_t)src * D + d];
}

// ---------------- launcher ----------------

extern "C" void kernel_launch(void* const* d_in, const int* in_sizes, int n_in,
                              void* d_out, int out_size, void* d_ws, size_t ws_size,
                              hipStream_t stream) {
    const float* norm_adj = (const float*)d_in[0];
    const float* user_emb = (const float*)d_in[1];
    const float* item_emb = (const float*)d_in[2];
    const int*   users     = (const int*)d_in[3];
    const int*   pos_items = (const int*)d_in[4];
    const int*   neg_items = (const int*)d_in[5];
    // d_in[6] = mask, unused

    float* out = (float*)d_out;

    // workspace layout (bytes)
    char* ws = (char*)d_ws;
    float* flagf = (float*)ws;                                   // NTOT floats
    float* bcat  = (float*)(ws + (size_t)NTOT * 4);              // NTOT*128 floats
    float* light = (float*)(ws + (size_t)NTOT * 4
                               + (size_t)NTOT * 128 * 4);        // NTOT*64 floats

    zero_flag_kernel<<<(NTOT + 255) / 256, 256, 0, stream>>>(flagf);
    scatter_flag_kernel<<<(BATCH + 1 + 255) / 256, 256, 0, stream>>>(users, pos_items, flagf);
    build_bcat_kernel<<<(NTOT * D + 255) / 256, 256, 0, stream>>>(user_emb, item_emb, flagf, bcat);
    gemm_light_kernel<<<NTOT / 16, 256, 0, stream>>>(norm_adj, bcat, flagf,
                                                     user_emb, item_emb, light);
    int gather_elems = (2 + NNEG) * BATCH * D;
    gather_out_kernel<<<(gather_elems + 255) / 256, 256, 0, stream>>>(
        light, users, pos_items, neg_items, out);
}